// DeconvCG_31997506355774
// MI455X (gfx1250) — compile-verified
//
#include <hip/hip_runtime.h>

// Bilateral grid (splat -> separable 5-tap blur -> trilinear slice)
// 12 channels of 1024x1024, grid 129x129x17 (z padded to 20 for alignment).
// y/x blur: async global->LDS staging (ASYNCcnt) + LDS stencil.
// z-blur: f32 WMMA, OUT(16x16) = F(16x20) x IN(20x16), K=20 in 5 chunks.

#define HH   1024
#define WW   1024
#define NPIX (HH * WW)
#define NBm1 15.0f
#define GH   129
#define GW   129
#define GZ   17
#define GZP  20
#define NCELL (GH * GW)         // 16641
#define GRIDF (NCELL * GZP)     // 332820 floats per grid buffer

typedef float v2f __attribute__((ext_vector_type(2)));
typedef float v8f __attribute__((ext_vector_type(8)));

#if defined(__has_builtin)
#if __has_builtin(__builtin_amdgcn_global_load_async_to_lds_b32) && \
    __has_builtin(__builtin_amdgcn_s_wait_asynccnt)
#define HAVE_ASYNC_LDS 1
#endif
#endif

typedef __attribute__((address_space(1))) int gas_int;   // global addrspace
typedef __attribute__((address_space(3))) int las_int;   // LDS addrspace

// ---------------------------------------------------------------- zero
__global__ void zero_kernel(float4* __restrict__ p, int n4) {
    int i = blockIdx.x * blockDim.x + threadIdx.x;
    if (i < n4) p[i] = make_float4(0.f, 0.f, 0.f, 0.f);
}

// ---------------------------------------------------------------- splat
__global__ void splat_kernel(const float* __restrict__ img,
                             float* __restrict__ val, float* __restrict__ wt) {
    int p = blockIdx.x * blockDim.x + threadIdx.x;
    if (p >= NPIX) return;
    int y = p >> 10;
    int x = p & 1023;
    float v = img[p];
    // round-half-to-even matches jnp.round (v_rndne_f32)
    int gy = (int)rintf((float)y * 0.125f);
    int gx = (int)rintf((float)x * 0.125f);
    float r = rintf(v * NBm1);
    r = fminf(fmaxf(r, 0.0f), NBm1);
    int gz = (int)r;
    int idx = (gy * GW + gx) * GZP + gz;
    unsafeAtomicAdd(&val[idx], v);
    unsafeAtomicAdd(&wt[idx], 1.0f);
}

// ---------------------------------------------------------------- blur y / x
// One thread per PADDED grid element (contiguous block footprint).
// Each block stages 5 tap-rows (contiguous 256-float segments) of val & wt
// into LDS with async global->LDS loads, then computes the stencil from LDS.
__global__ __launch_bounds__(256)
void blur_yx_kernel(const float* __restrict__ srcV, const float* __restrict__ srcW,
                    float* __restrict__ dstV, float* __restrict__ dstW,
                    const float* __restrict__ fs, int axis) {
    __shared__ float sV[5][256];
    __shared__ float sW[5][256];
    int tid = threadIdx.x;
    int t0  = blockIdx.x * 256;
    int stride = (axis == 0) ? (GW * GZP) : GZP;

#pragma unroll
    for (int r = 0; r < 5; ++r) {
        int off = t0 + (r - 2) * stride + tid;
        off = max(0, min(off, GRIDF - 1));     // clamp into buffer; OOB taps masked below
#if defined(HAVE_ASYNC_LDS)
        __builtin_amdgcn_global_load_async_to_lds_b32(
            (gas_int*)(srcV + off), (las_int*)&sV[r][tid], 0, 0);
        __builtin_amdgcn_global_load_async_to_lds_b32(
            (gas_int*)(srcW + off), (las_int*)&sW[r][tid], 0, 0);
#else
        sV[r][tid] = srcV[off];
        sW[r][tid] = srcW[off];
#endif
    }
#if defined(HAVE_ASYNC_LDS)
    __builtin_amdgcn_s_wait_asynccnt(0);
#endif
    __syncthreads();

    int t = t0 + tid;
    if (t >= GRIDF) return;
    int cell = t / GZP;
    int z = t - cell * GZP;
    if (z >= GZ) return;                       // padding lanes only helped staging
    int y = cell / GW;
    int x = cell - y * GW;
    int coord = (axis == 0) ? y : x;

    float k0 = fs[0], k1 = fs[1], k2 = fs[2], k3 = fs[3], k4 = fs[4];
    float aV = 0.f, aW = 0.f;
#pragma unroll
    for (int tap = 0; tap < 5; ++tap) {
        int c2 = coord + tap - 2;
        if (c2 >= 0 && c2 < GH) {              // GH == GW == 129; zero-pad semantics
            float kk = (tap == 0) ? k0 : (tap == 1) ? k1 : (tap == 2) ? k2
                     : (tap == 3) ? k3 : k4;
            aV += kk * sV[tap][tid];
            aW += kk * sW[tap][tid];
        }
    }
    dstV[t] = aV;
    dstW[t] = aW;
}

// ---------------------------------------------------------------- blur z (WMMA)
// F[m][k] = fr[k-m+2] if (0 <= k-m+2 < 5 && k < 17) else 0   (zero-pad folded in)
__device__ __forceinline__ float fcoef(int m, int k,
                                       float f0, float f1, float f2, float f3, float f4) {
    int d = k - m + 2;
    float r = 0.0f;
    r = (d == 0) ? f0 : r;
    r = (d == 1) ? f1 : r;
    r = (d == 2) ? f2 : r;
    r = (d == 3) ? f3 : r;
    r = (d == 4) ? f4 : r;
    r = (k < GZ) ? r : 0.0f;
    return r;
}

__global__ void blur_z_wmma_kernel(const float* __restrict__ srcV, const float* __restrict__ srcW,
                                   float* __restrict__ dstV, float* __restrict__ dstW,
                                   const float* __restrict__ fr) {
    const int NT = (NCELL + 15) / 16;                 // 1041 cell tiles
    int wave = (int)((blockIdx.x * blockDim.x + threadIdx.x) >> 5);
    int lane = (int)(threadIdx.x & 31);
    if (wave >= NT * 2) return;                       // wave-uniform exit: EXEC stays full
    int tile  = wave >> 1;
    int which = wave & 1;
    const float* __restrict__ src = which ? srcW : srcV;
    float* __restrict__ dst       = which ? dstW : dstV;

    int h = lane >> 4;                                 // half-wave
    int n = lane & 15;                                 // column / A-matrix row
    int cell  = tile * 16 + n;
    int cellc = (cell < NCELL) ? cell : (NCELL - 1);   // clamp loads, predicate stores
    int sbase = cellc * GZP;

    float f0 = fr[0], f1 = fr[1], f2 = fr[2], f3 = fr[3], f4 = fr[4];

    v8f c = {0.f, 0.f, 0.f, 0.f, 0.f, 0.f, 0.f, 0.f};
#pragma unroll
    for (int kb = 0; kb < 5; ++kb) {                   // K = 20 in chunks of 4
        int k0 = kb * 4 + h * 2;
        v2f a;
        a.x = fcoef(n, k0,     f0, f1, f2, f3, f4);
        a.y = fcoef(n, k0 + 1, f0, f1, f2, f3, f4);
        v2f b = *(const v2f*)(src + sbase + k0);       // 8B-aligned (even offsets)
        c = __builtin_amdgcn_wmma_f32_16x16x4_f32(
                /*neg_a=*/false, a, /*neg_b=*/false, b,
                /*c_mod=*/(short)0, c, /*reuse_a=*/false, /*reuse_b=*/false);
    }

    if (cell < NCELL) {
        int dbase = cell * GZP;
#pragma unroll
        for (int r = 0; r < 8; ++r)                    // D: lanes0-15 M=r, lanes16-31 M=r+8
            dst[dbase + r + 8 * h] = c[r];
        if (h == 0) {                                  // output row z = 16 (scalar)
            float o16 = f0 * src[sbase + 14] + f1 * src[sbase + 15] + f2 * src[sbase + 16];
            dst[dbase + 16] = o16;
        }
    }
}

// ---------------------------------------------------------------- slice
__global__ void slice_kernel(const float* __restrict__ img,
                             const float* __restrict__ val, const float* __restrict__ wt,
                             float* __restrict__ out) {
    int p = blockIdx.x * blockDim.x + threadIdx.x;
    if (p >= NPIX) return;
    int y = p >> 10;
    int x = p & 1023;
    float v = img[p];

    int y0 = y >> 3;
    int y1 = min(y0 + 1, GH - 1);
    float ty = (float)(y & 7) * 0.125f;
    int x0 = x >> 3;
    int x1 = min(x0 + 1, GW - 1);
    float tx = (float)(x & 7) * 0.125f;
    float fz = fminf(fmaxf(v * NBm1, 0.0f), NBm1);
    int z0 = (int)fz;
    int z1 = min(z0 + 1, GZ - 1);
    float tz = fz - (float)z0;

    int b00 = (y0 * GW + x0) * GZP;
    int b01 = (y0 * GW + x1) * GZP;
    int b10 = (y1 * GW + x0) * GZP;
    int b11 = (y1 * GW + x1) * GZP;
    float wy0 = 1.f - ty, wy1 = ty, wx0 = 1.f - tx, wx1 = tx, wz0 = 1.f - tz, wz1 = tz;

    float av = 0.f, aw = 0.f;
#define CORNER(b, wyx)                                               \
    {                                                                \
        float w0 = (wyx) * wz0, w1 = (wyx) * wz1;                    \
        av += w0 * val[(b) + z0] + w1 * val[(b) + z1];               \
        aw += w0 * wt[(b) + z0] + w1 * wt[(b) + z1];                 \
    }
    CORNER(b00, wy0 * wx0)
    CORNER(b01, wy0 * wx1)
    CORNER(b10, wy1 * wx0)
    CORNER(b11, wy1 * wx1)
#undef CORNER

    out[p] = av / (aw + 1e-8f);
}

// ---------------------------------------------------------------- launch
extern "C" void kernel_launch(void* const* d_in, const int* in_sizes, int n_in,
                              void* d_out, int out_size, void* d_ws, size_t ws_size,
                              hipStream_t stream) {
    (void)in_sizes; (void)n_in; (void)out_size; (void)ws_size;
    const float* imgs = (const float*)d_in[0];   // [4,3,1024,1024]
    const float* fs   = (const float*)d_in[2];   // 5 taps
    const float* fr   = (const float*)d_in[3];   // 5 taps
    float* out = (float*)d_out;

    float* ws = (float*)d_ws;                    // 4 grid buffers, 5.33 MB total
    float* Av = ws;
    float* Aw = ws + (size_t)GRIDF;
    float* Bv = ws + (size_t)2 * GRIDF;
    float* Bw = ws + (size_t)3 * GRIDF;

    const int BLK = 256;
    const int zeroN4   = GRIDF;                  // 4*GRIDF floats / 4 per float4
    const int blurBlks = (GRIDF + BLK - 1) / BLK;
    const int zWaves   = ((NCELL + 15) / 16) * 2;
    const int zThreads = zWaves * 32;

    for (int c = 0; c < 12; ++c) {
        const float* img = imgs + (size_t)c * NPIX;
        float* o         = out  + (size_t)c * NPIX;
        zero_kernel <<<(zeroN4 + BLK - 1) / BLK, BLK, 0, stream>>>((float4*)ws, zeroN4);
        splat_kernel<<<(NPIX + BLK - 1) / BLK, BLK, 0, stream>>>(img, Av, Aw);
        blur_yx_kernel<<<blurBlks, BLK, 0, stream>>>(Av, Aw, Bv, Bw, fs, 0);
        blur_yx_kernel<<<blurBlks, BLK, 0, stream>>>(Bv, Bw, Av, Aw, fs, 1);
        blur_z_wmma_kernel<<<(zThreads + BLK - 1) / BLK, BLK, 0, stream>>>(Av, Aw, Bv, Bw, fr);
        slice_kernel<<<(NPIX + BLK - 1) / BLK, BLK, 0, stream>>>(img, Bv, Bw, o);
    }
}